// AttentionGNN_88098369175681
// MI455X (gfx1250) — compile-verified
//
#include <hip/hip_runtime.h>
#include <math.h>

#define IN_F 256
#define HID_F 128
#define OUT_F 64
#define NEG_SLOPE 0.2f
#define GAT_EPS 1e-16f

typedef float v2f __attribute__((ext_vector_type(2)));
typedef float v8f __attribute__((ext_vector_type(8)));

// ---- monotone float<->uint encoding so atomicMax(uint) == float max ----
__device__ __forceinline__ unsigned enc_f32(float f) {
    unsigned u = __float_as_uint(f);
    return (u & 0x80000000u) ? ~u : (u | 0x80000000u);
}
__device__ __forceinline__ float dec_f32(unsigned u) {
    return (u & 0x80000000u) ? __uint_as_float(u ^ 0x80000000u)
                             : __uint_as_float(~u);
}

// ---------------------------------------------------------------------------
// out[M,N] = act(A[M,K] @ W[K,N] + bias[N]); one wave per 16x16 C tile,
// V_WMMA_F32_16X16X4_F32 along K. blockDim.x = 32*(N/16).
// A layout (ISA 7.12.2): lanes 0-15 -> M rows, VGPR0=K+0/K+2, VGPR1=K+1/K+3
// selected by lane[4]. B mirrored with N = lane&15.
// ---------------------------------------------------------------------------
template <int K, int N, bool ELU>
__global__ void gat_gemm_wmma(const float* __restrict__ A,
                              const float* __restrict__ W,
                              const float* __restrict__ bias,
                              float* __restrict__ out, int M) {
    const int lane = threadIdx.x & 31;
    const int wave = threadIdx.x >> 5;
    const int m0 = blockIdx.x * 16;
    const int n0 = wave * 16;
    int row = m0 + (lane & 15);
    if (row >= M) row = M - 1;              // keep EXEC all-ones for WMMA
    const int kl = (lane >> 4) * 2;         // 0 or 2
    const int n = n0 + (lane & 15);

    v8f acc = {};
    const float* __restrict__ arow = A + (size_t)row * K;
#pragma unroll 4
    for (int k = 0; k < K; k += 4) {
        const float2 av = *(const float2*)(arow + k + kl);
        v2f a;
        a.x = av.x;
        a.y = av.y;
        v2f b;
        b.x = W[(size_t)(k + kl) * N + n];
        b.y = W[(size_t)(k + kl + 1) * N + n];
        acc = __builtin_amdgcn_wmma_f32_16x16x4_f32(false, a, false, b,
                                                    (short)0, acc, false, false);
    }
    const float bn = bias[n];
#pragma unroll
    for (int r = 0; r < 8; ++r) {
        const int m = m0 + r + ((lane >> 4) << 3);
        if (m < M) {
            float v = acc[r] + bn;
            if (ELU) v = (v > 0.f) ? v : (expf(v) - 1.f);
            out[(size_t)m * N + n] = v;
        }
    }
}

// ---------------------------------------------------------------------------
// Per-node attention dot products: ds[i] = h_i . aw[:D], dt[i] = h_i . aw[D:]
// One wave per node.
// ---------------------------------------------------------------------------
template <int D>
__global__ void node_att_dots(const float* __restrict__ h,
                              const float* __restrict__ aw,
                              float* __restrict__ ds, float* __restrict__ dt,
                              int M) {
    const int lane = threadIdx.x & 31;
    const int node = blockIdx.x * (blockDim.x >> 5) + (threadIdx.x >> 5);
    if (node >= M) return;
    const float* __restrict__ hv = h + (size_t)node * D;
    float s = 0.f, t = 0.f;
#pragma unroll
    for (int j = lane; j < D; j += 32) {
        const float v = hv[j];
        s += v * aw[j];
        t += v * aw[D + j];
    }
    for (int o = 16; o > 0; o >>= 1) {
        s += __shfl_xor(s, o, 32);
        t += __shfl_xor(t, o, 32);
    }
    if (lane == 0) {
        ds[node] = s;
        dt[node] = t;
    }
}

// Pass A: alpha = leaky_relu(ds[src]+dt[tgt]+b); segment max by tgt.
__global__ void edge_alpha_max(const int* __restrict__ src,
                               const int* __restrict__ tgt,
                               const float* __restrict__ ds,
                               const float* __restrict__ dt,
                               const float* __restrict__ ab,
                               float* __restrict__ alpha,
                               unsigned* __restrict__ menc, int E) {
    const int e = blockIdx.x * blockDim.x + threadIdx.x;
    if (e >= E) return;
    float a = ds[src[e]] + dt[tgt[e]] + ab[0];
    a = (a >= 0.f) ? a : NEG_SLOPE * a;
    alpha[e] = a;
    atomicMax(&menc[tgt[e]], enc_f32(a));
}

// Pass B: e = exp(alpha - m[tgt]); segment sum by tgt (alpha overwritten w/ e).
__global__ void edge_exp_sum(const int* __restrict__ tgt,
                             const unsigned* __restrict__ menc,
                             float* __restrict__ alpha,
                             float* __restrict__ ssum, int E) {
    const int e = blockIdx.x * blockDim.x + threadIdx.x;
    if (e >= E) return;
    const int t = tgt[e];
    const float ex = expf(alpha[e] - dec_f32(menc[t]));
    alpha[e] = ex;
    atomicAdd(&ssum[t], ex);
}

// Pass C: agg[tgt] += h[src] * (e / (s[tgt]+eps)); one wave per edge.
template <int D>
__global__ void edge_scatter_msgs(const int* __restrict__ src,
                                  const int* __restrict__ tgt,
                                  const float* __restrict__ ew,
                                  const float* __restrict__ ssum,
                                  const float* __restrict__ h,
                                  float* __restrict__ agg, int E) {
    const int lane = threadIdx.x & 31;
    const int e = blockIdx.x * (blockDim.x >> 5) + (threadIdx.x >> 5);
    if (e >= E) return;
    const int s = src[e], t = tgt[e];
    const float w = ew[e] / (ssum[t] + GAT_EPS);
    const float* __restrict__ hv = h + (size_t)s * D;
    float* __restrict__ av = agg + (size_t)t * D;
#pragma unroll
    for (int j = lane; j < D; j += 32) atomicAdd(&av[j], hv[j] * w);
}

// log_softmax over 64 classes; one wave per node (2 elems per lane).
__global__ void log_softmax64(const float* __restrict__ o,
                              float* __restrict__ out, int M) {
    const int lane = threadIdx.x & 31;
    const int node = blockIdx.x * (blockDim.x >> 5) + (threadIdx.x >> 5);
    if (node >= M) return;
    const float* __restrict__ v = o + (size_t)node * OUT_F;
    const float v0 = v[lane], v1 = v[lane + 32];
    float mx = fmaxf(v0, v1);
    for (int off = 16; off > 0; off >>= 1) mx = fmaxf(mx, __shfl_xor(mx, off, 32));
    float s = expf(v0 - mx) + expf(v1 - mx);
    for (int off = 16; off > 0; off >>= 1) s += __shfl_xor(s, off, 32);
    const float l = mx + logf(s);
    out[(size_t)node * OUT_F + lane] = v0 - l;
    out[(size_t)node * OUT_F + lane + 32] = v1 - l;
}

__global__ void zero_f32(float* __restrict__ p, size_t n) {
    size_t i = (size_t)blockIdx.x * blockDim.x + threadIdx.x;
    const size_t stride = (size_t)gridDim.x * blockDim.x;
    for (; i < n; i += stride) p[i] = 0.f;
}

extern "C" void kernel_launch(void* const* d_in, const int* in_sizes, int n_in,
                              void* d_out, int out_size, void* d_ws,
                              size_t ws_size, hipStream_t stream) {
    const float* x = (const float*)d_in[0];
    const int* ei = (const int*)d_in[1];
    const float* W1 = (const float*)d_in[2];
    const float* b1 = (const float*)d_in[3];
    const float* a1w = (const float*)d_in[4];
    const float* a1b = (const float*)d_in[5];
    const float* W2 = (const float*)d_in[6];
    const float* b2 = (const float*)d_in[7];
    const float* a2w = (const float*)d_in[8];
    const float* a2b = (const float*)d_in[9];
    float* outp = (float*)d_out;

    const int Nn = in_sizes[0] / IN_F;   // 50000
    const int E = in_sizes[1] / 2;       // 800000
    const int* src = ei;
    const int* tgt = ei + E;

    // ---- workspace layout ----
    float* ws = (float*)d_ws;
    const size_t nN = (size_t)Nn;
    float* h = ws;                         // nN*128
    float* hagg = h + nN * HID_F;          // nN*128
    float* o = hagg + nN * HID_F;          // nN*64
    float* oagg = o + nN * OUT_F;          // nN*64
    float* ds1 = oagg + nN * OUT_F;        // nN
    float* dt1 = ds1 + nN;                 // nN
    float* s1 = dt1 + nN;                  // nN
    unsigned* m1 = (unsigned*)(s1 + nN);   // nN
    float* ds2 = (float*)(m1 + nN);        // nN
    float* dt2 = ds2 + nN;                 // nN
    float* s2 = dt2 + nN;                  // nN
    unsigned* m2 = (unsigned*)(s2 + nN);   // nN
    float* alpha = (float*)(m2 + nN);      // E  (reused for both layers)

    const int zb = 2048, zt = 256;
    // zero accumulators (every call: deterministic across graph replays)
    zero_f32<<<zb, zt, 0, stream>>>(hagg, nN * HID_F);
    zero_f32<<<zb, zt, 0, stream>>>(oagg, nN * OUT_F);
    zero_f32<<<zb, zt, 0, stream>>>(s1, nN * 2);   // s1 + m1 (enc 0 ~ -inf)
    zero_f32<<<zb, zt, 0, stream>>>(s2, nN * 2);   // s2 + m2

    const int mtiles = (Nn + 15) / 16;
    const int waveBlocks = (Nn + 7) / 8;   // 8 waves of 32 per 256-thread block
    const int edgeBlocks = (E + 255) / 256;
    const int edgeWaveBlocks = (E + 7) / 8;

    // ---- layer 1: h = elu(x @ W1 + b1) ----
    gat_gemm_wmma<IN_F, HID_F, true>
        <<<mtiles, 32 * (HID_F / 16), 0, stream>>>(x, W1, b1, h, Nn);
    node_att_dots<HID_F><<<waveBlocks, 256, 0, stream>>>(h, a1w, ds1, dt1, Nn);
    edge_alpha_max<<<edgeBlocks, 256, 0, stream>>>(src, tgt, ds1, dt1, a1b,
                                                   alpha, m1, E);
    edge_exp_sum<<<edgeBlocks, 256, 0, stream>>>(tgt, m1, alpha, s1, E);
    edge_scatter_msgs<HID_F>
        <<<edgeWaveBlocks, 256, 0, stream>>>(src, tgt, alpha, s1, h, hagg, E);

    // ---- layer 2: o = hagg @ W2 + b2 ----
    gat_gemm_wmma<HID_F, OUT_F, false>
        <<<mtiles, 32 * (OUT_F / 16), 0, stream>>>(hagg, W2, b2, o, Nn);
    node_att_dots<OUT_F><<<waveBlocks, 256, 0, stream>>>(o, a2w, ds2, dt2, Nn);
    edge_alpha_max<<<edgeBlocks, 256, 0, stream>>>(src, tgt, ds2, dt2, a2b,
                                                   alpha, m2, E);
    edge_exp_sum<<<edgeBlocks, 256, 0, stream>>>(tgt, m2, alpha, s2, E);
    edge_scatter_msgs<OUT_F>
        <<<edgeWaveBlocks, 256, 0, stream>>>(src, tgt, alpha, s2, o, oagg, E);

    // ---- log_softmax ----
    log_softmax64<<<waveBlocks, 256, 0, stream>>>(oagg, outp, Nn);
    (void)n_in; (void)out_size; (void)ws_size;
}